// LyotFilter_21457656611281
// MI455X (gfx1250) — compile-verified
//
#include <hip/hip_runtime.h>
#include <math.h>

typedef float v2f __attribute__((ext_vector_type(2)));
typedef float v8f __attribute__((ext_vector_type(8)));

#define IN_DIM   200
#define OUT_DIM  64
#define WSTRIDE  210           // floats; 840B pitch: 8B-aligned, conflict-free columns
#define ROWS_PER_BLOCK 128     // 8 waves x 16 rows; each wave does all 64 output cols
#define KCHUNK   40            // 10 k-steps per outer iteration (200 = 5 chunks)

__global__ __launch_bounds__(256) void lyot_wmma_gemm(const float* __restrict__ x,
                                                      const float* __restrict__ weight_,
                                                      float* __restrict__ out,
                                                      int nrows) {
    __shared__ float wlds[OUT_DIM * WSTRIDE];   // w_norm^T : [o][k], 53760 B

    const int tid = threadIdx.x;

    // ---- per-block weight construction: w_norm[k][o], stored transposed ----
    if (tid < OUT_DIM) {
        const int o = tid;
        const float num = -0.06283185307179586f / weight_[o];  // 2*pi*(-0.01)/w_o
        float sum = 0.0f;
        for (int k = 0; k < IN_DIM; ++k) {
            // kept-band index: skip n in [103,107],[149,162],{219}
            const int n = k + (k < 103 ? 0 : (k < 144 ? 5 : 19));
            const float band = (float)(400.0 + 2100.0 * (double)n / 220.0);
            const float phase = num / (band * 1e-6f);
            const float val = 0.5f - 0.5f * cosf(phase);
            wlds[o * WSTRIDE + k] = val;
            sum += val;
        }
        const float inv = 1.0f / sum;
        for (int k = 0; k < IN_DIM; ++k) {
            const float v = wlds[o * WSTRIDE + k];
            wlds[o * WSTRIDE + k] = fmaxf(v, 0.0f) * inv;
        }
    }
    __syncthreads();

    // ---- wave tiling: 16 rows x 64 cols per wave (x read exactly once) ----
    const int lane = tid & 31;
    const int wv   = tid >> 5;             // 0..7
    const int l16  = lane & 15;
    const int khi  = (lane >> 4) << 1;     // lanes 16-31 carry K = k0+2, k0+3

    const int rb = blockIdx.x * ROWS_PER_BLOCK + wv * 16;

    long long r = rb + l16;  if (r >= nrows) r = nrows - 1;   // clamp loads only
    const float* ap = x + r * IN_DIM + khi;
    const float* bp = &wlds[l16 * WSTRIDE + khi];   // o-tile t at +t*16*WSTRIDE

    v8f acc0 = {}, acc1 = {}, acc2 = {}, acc3 = {};

    #pragma unroll 1
    for (int kc = 0; kc < IN_DIM; kc += KCHUNK) {
        __builtin_prefetch(ap + 2 * KCHUNK, 0, 3);   // global_prefetch_b8, ~320B ahead
        #pragma unroll
        for (int k0 = 0; k0 < KCHUNK; k0 += 4) {
            const v2f a  = *(const v2f*)(ap + k0);
            const v2f b0 = *(const v2f*)(bp + k0);
            const v2f b1 = *(const v2f*)(bp + 16 * WSTRIDE + k0);
            const v2f b2 = *(const v2f*)(bp + 32 * WSTRIDE + k0);
            const v2f b3 = *(const v2f*)(bp + 48 * WSTRIDE + k0);
            acc0 = __builtin_amdgcn_wmma_f32_16x16x4_f32(false, a, false, b0,
                                                         (short)0, acc0, false, false);
            acc1 = __builtin_amdgcn_wmma_f32_16x16x4_f32(false, a, false, b1,
                                                         (short)0, acc1, false, false);
            acc2 = __builtin_amdgcn_wmma_f32_16x16x4_f32(false, a, false, b2,
                                                         (short)0, acc2, false, false);
            acc3 = __builtin_amdgcn_wmma_f32_16x16x4_f32(false, a, false, b3,
                                                         (short)0, acc3, false, false);
        }
        ap += KCHUNK;
        bp += KCHUNK;
    }

    // ---- store: D layout VGPR i <-> (M=i | M=8+i), N = lane%16 ----
    const int rlo = (lane >> 4) * 8;
    float* op = out + (long long)(rb + rlo) * OUT_DIM + l16;

    if (rb + 16 <= nrows) {                 // wave-uniform fast path (nrows % 16 == 0)
        #pragma unroll
        for (int i = 0; i < 8; ++i) {
            op[i * OUT_DIM +  0] = acc0[i];
            op[i * OUT_DIM + 16] = acc1[i];
            op[i * OUT_DIM + 32] = acc2[i];
            op[i * OUT_DIM + 48] = acc3[i];
        }
    } else {                                // ragged tail (unused at N=500000)
        #pragma unroll
        for (int i = 0; i < 8; ++i) {
            if (rb + rlo + i < nrows) {
                op[i * OUT_DIM +  0] = acc0[i];
                op[i * OUT_DIM + 16] = acc1[i];
                op[i * OUT_DIM + 32] = acc2[i];
                op[i * OUT_DIM + 48] = acc3[i];
            }
        }
    }
}

extern "C" void kernel_launch(void* const* d_in, const int* in_sizes, int n_in,
                              void* d_out, int out_size, void* d_ws, size_t ws_size,
                              hipStream_t stream) {
    const float* x  = (const float*)d_in[0];
    const float* w_ = (const float*)d_in[1];
    float* out = (float*)d_out;
    const int nrows = in_sizes[0] / IN_DIM;
    const int grid  = (nrows + ROWS_PER_BLOCK - 1) / ROWS_PER_BLOCK;
    hipLaunchKernelGGL(lyot_wmma_gemm, dim3(grid), dim3(256), 0, stream,
                       x, w_, out, nrows);
}